// GCNConv_60078002536567
// MI455X (gfx1250) — compile-verified
//
#include <hip/hip_runtime.h>

typedef float v2f __attribute__((ext_vector_type(2)));
typedef float v4f __attribute__((ext_vector_type(4)));
typedef float v8f __attribute__((ext_vector_type(8)));

#define D_FEAT      128
#define TILE_ROWS   128
#define LDS_STRIDE  132   // 128 + 4 pad: wmma frag b64 loads hit 64 distinct banks

// Fused GCN layer: out = Dd * (A_csr @ (Ds * X)) @ W
// One block = 128 dst nodes. Stage W^T + aggregated rows in LDS (132 KB,
// only possible on CDNA5's 320KB WGP LDS), then WMMA f32 16x16x4 GEMM.
__global__ __launch_bounds__(256)
void gcn_fused_kernel(const int* __restrict__ edge_ptr,
                      const int* __restrict__ src_edges,
                      const float* __restrict__ src_norm,
                      const float* __restrict__ dst_norm,
                      const float* __restrict__ X,
                      const float* __restrict__ W,
                      float* __restrict__ out,
                      int n_nodes)
{
    extern __shared__ float smem[];
    float* sA  = smem;                               // [TILE_ROWS][LDS_STRIDE]
    float* sWt = smem + TILE_ROWS * LDS_STRIDE;      // [128][LDS_STRIDE], W transposed

    const int tid     = threadIdx.x;
    const int lane    = tid & 31;
    const int wave    = tid >> 5;                    // 0..7
    const int rowBase = blockIdx.x * TILE_ROWS;

    // ---- Stage W^T into LDS (W is 64KB, L2-resident; staging cost is tiny) ----
    for (int i = tid; i < (D_FEAT * D_FEAT) / 4; i += 256) {
        int k  = i >> 5;            // input-feature index 0..127
        int c4 = (i & 31) << 2;     // output col 0,4,...,124
        v4f w4 = *(const v4f*)(W + k * D_FEAT + c4);
        sWt[(c4 + 0) * LDS_STRIDE + k] = w4.x;
        sWt[(c4 + 1) * LDS_STRIDE + k] = w4.y;
        sWt[(c4 + 2) * LDS_STRIDE + k] = w4.z;
        sWt[(c4 + 3) * LDS_STRIDE + k] = w4.w;
    }

    // ---- Stage 1: gather + normalize + aggregate. One wave -> 16 nodes. ----
    // Each lane owns 4 consecutive feature columns (float4 loads, L2-served).
    const int colBase = lane << 2;
    for (int j = 0; j < 16; ++j) {
        int r    = wave * 16 + j;
        int node = rowBase + r;
        v4f acc = {0.f, 0.f, 0.f, 0.f};
        if (node < n_nodes) {
            int nu = __builtin_amdgcn_readfirstlane(node);   // force SGPR -> SMEM loads
            int e0 = edge_ptr[nu];
            int e1 = edge_ptr[nu + 1];
            for (int e = e0; e < e1; ++e) {
                int   s   = src_edges[e];    // uniform -> s_load
                float wsc = src_norm[s];     // uniform -> s_load
                v4f x4 = *(const v4f*)(X + s * D_FEAT + colBase);
                acc.x = fmaf(wsc, x4.x, acc.x);
                acc.y = fmaf(wsc, x4.y, acc.y);
                acc.z = fmaf(wsc, x4.z, acc.z);
                acc.w = fmaf(wsc, x4.w, acc.w);
            }
            float dn = dst_norm[nu];
            acc.x *= dn; acc.y *= dn; acc.z *= dn; acc.w *= dn;
        }
        // out-of-range rows write zeros so stage-2 WMMA reads clean data
        *(v4f*)(sA + r * LDS_STRIDE + colBase) = acc;
    }

    __syncthreads();

    // ---- Stage 2: each wave computes a 16x128 output stripe with fp32 WMMA ----
    // A frag (16x4 f32): lane L holds A[L&15][4k + kb], kb = (L>>4)*2 -> b64 from sA
    // B frag (4x16 f32): lane L holds B[4k+kb][ct*16 + (L&15)] = sWt[col][4k+kb] -> b64
    const int mrow = lane & 15;
    const int kb   = (lane >> 4) << 1;     // 0 or 2
    const float* aRow = sA + (wave * 16 + mrow) * LDS_STRIDE + kb;

    v8f c[8];
#pragma unroll
    for (int ct = 0; ct < 8; ++ct) c[ct] = (v8f){0.f,0.f,0.f,0.f,0.f,0.f,0.f,0.f};

    for (int k4 = 0; k4 < D_FEAT / 4; ++k4) {
        v2f a = *(const v2f*)(aRow + 4 * k4);
#pragma unroll
        for (int ct = 0; ct < 8; ++ct) {
            v2f b = *(const v2f*)(sWt + (ct * 16 + mrow) * LDS_STRIDE + 4 * k4 + kb);
            // (neg_a, A, neg_b, B, c_mod, C, reuse_a, reuse_b)
            c[ct] = __builtin_amdgcn_wmma_f32_16x16x4_f32(
                        false, a, false, b, (short)0, c[ct], false, false);
        }
    }

    // ---- Store: lanes 0..15 cover 16 consecutive cols -> 64B coalesced stores ----
    const int mhalf = (lane >> 4) << 3;    // 0 or 8
#pragma unroll
    for (int ct = 0; ct < 8; ++ct) {
#pragma unroll
        for (int r = 0; r < 8; ++r) {
            int row = rowBase + wave * 16 + mhalf + r;
            if (row < n_nodes) {
                out[row * D_FEAT + (ct * 16 + mrow)] = c[ct][r];
            }
        }
    }
}

extern "C" void kernel_launch(void* const* d_in, const int* in_sizes, int n_in,
                              void* d_out, int out_size, void* d_ws, size_t ws_size,
                              hipStream_t stream) {
    const int*   edge_ptr  = (const int*)  d_in[0];
    const int*   src_edges = (const int*)  d_in[1];
    const float* src_norm  = (const float*)d_in[2];
    const float* dst_norm  = (const float*)d_in[3];
    // d_in[4] = dst_nodes scalar (device); recover N from host-visible sizes
    const float* input_feat = (const float*)d_in[5];
    const float* weight     = (const float*)d_in[6];
    float* out = (float*)d_out;

    const int n_nodes = in_sizes[3];                       // dst_norm_degs has N entries
    const int grid    = (n_nodes + TILE_ROWS - 1) / TILE_ROWS;
    const size_t shmem = 2u * TILE_ROWS * LDS_STRIDE * sizeof(float);  // 132 KB

    // Allow >64KB dynamic LDS (CDNA5 WGP has 320KB). Idempotent host-side
    // attribute set, not a stream op -> capture-safe; called unconditionally
    // (no static state) so kernel_launch stays deterministic.
    (void)hipFuncSetAttribute((const void*)gcn_fused_kernel,
                              hipFuncAttributeMaxDynamicSharedMemorySize,
                              (int)shmem);

    gcn_fused_kernel<<<grid, 256, shmem, stream>>>(
        edge_ptr, src_edges, src_norm, dst_norm, input_feat, weight, out, n_nodes);
}